// FSQuantizer_36807869727046
// MI455X (gfx1250) — compile-verified
//
#include <hip/hip_runtime.h>

// FSQ quantizer for MI455X (gfx1250, wave32).
// Memory-bound streaming kernel: 192 MiB total traffic -> ~8.6 us floor @ 23.3 TB/s.
// No GEMM structure => WMMA not applicable; optimize the data-movement path:
//  - b128 non-temporal loads/stores (th:TH_*_NT) so the 192 MiB stream doesn't
//    evict the 1.5 MB histogram from the 192 MB L2
//  - no-return global_atomic_add_u32 for the 390,625-bin histogram (too big for LDS)
//  - NO tanh evaluation: round(2*tanh(x)) is a monotone step function of x, so the
//    quantizer reduces to two |x| threshold compares per element (atanh of the cell
//    edges), cutting VALU ~4x and keeping the kernel firmly under the HBM roofline.

typedef float f32x4 __attribute__((ext_vector_type(4)));

#define NPIX   (32 * 256 * 256)   // 2,097,152 pixels
#define NELEM  (NPIX * 8)         // 16,777,216 floats per [B,H,W,D] tensor
#define NBINS  390625             // 5^8

// Quantizer cell edges in x-space (tanh is monotone):
//   |zhat| >= 1  <=>  2*tanh(|x|) >  0.5  <=>  |x| >  atanh(0.25)   (rintf(0.5)  -> 0, ties-to-even)
//   |zhat| == 2  <=>  2*tanh(|x|) >= 1.5  <=>  |x| >= atanh(0.75)   (rintf(1.5)  -> 2, ties-to-even)
#define T1 0.25541281188299534f   // atanh(0.25)
#define T2 0.9729550745276566f    // atanh(0.75)

__global__ __launch_bounds__(256) void fsq_zero_usage(unsigned int* __restrict__ usage) {
    unsigned int i = blockIdx.x * 256u + threadIdx.x;
    if (i < NBINS) usage[i] = 0u;
}

// round(2*tanh(x)) without evaluating tanh: 2 compares + 2 selects + copysign.
static __device__ __forceinline__ float fsq_round(float x) {
    float ax = __builtin_fabsf(x);               // free as a source modifier
    float m  = (ax > T1) ? 1.0f : 0.0f;          // v_cmp + v_cndmask
    m        = (ax >= T2) ? 2.0f : m;            // v_cmp + v_cndmask
    return __builtin_copysignf(m, x);            // v_bfi; preserves -0.0 like jnp.round
}

__global__ __launch_bounds__(256) void fsq_main(const f32x4* __restrict__ x,
                                                f32x4* __restrict__ quant,
                                                f32x4* __restrict__ zhat,
                                                unsigned int* __restrict__ usage) {
    const unsigned int p = blockIdx.x * 256u + threadIdx.x;  // one pixel per lane
    const unsigned int v = p * 2u;                           // two float4 per pixel

    // 32 bytes per lane, streaming, non-temporal (global_load_b128 th:TH_LOAD_NT)
    f32x4 a = __builtin_nontemporal_load(x + v);
    f32x4 b = __builtin_nontemporal_load(x + v + 1);

    f32x4 ra, rb;
    ra.x = fsq_round(a.x);
    ra.y = fsq_round(a.y);
    ra.z = fsq_round(a.z);
    ra.w = fsq_round(a.w);
    rb.x = fsq_round(b.x);
    rb.y = fsq_round(b.y);
    rb.z = fsq_round(b.z);
    rb.w = fsq_round(b.w);

    // mixed-radix index: sum (zhat_i + 2) * 5^i   (digits 0..4, exact in fp32)
    unsigned int idx =
        (unsigned int)(ra.x + 2.0f) * 1u     +
        (unsigned int)(ra.y + 2.0f) * 5u     +
        (unsigned int)(ra.z + 2.0f) * 25u    +
        (unsigned int)(ra.w + 2.0f) * 125u   +
        (unsigned int)(rb.x + 2.0f) * 625u   +
        (unsigned int)(rb.y + 2.0f) * 3125u  +
        (unsigned int)(rb.z + 2.0f) * 15625u +
        (unsigned int)(rb.w + 2.0f) * 78125u;

    // quantized == zhat/2 exactly (z + (zhat - z) is exact by Sterbenz, then /2 is exact)
    __builtin_nontemporal_store(ra * 0.5f, quant + v);
    __builtin_nontemporal_store(rb * 0.5f, quant + v + 1);
    __builtin_nontemporal_store(ra, zhat + v);
    __builtin_nontemporal_store(rb, zhat + v + 1);

    // histogram: no-return global_atomic_add_u32, resolved in L2 atomic units.
    // Hottest bin gets ~18 of 2M updates -> negligible contention.
    atomicAdd(usage + idx, 1u);
}

extern "C" void kernel_launch(void* const* d_in, const int* in_sizes, int n_in,
                              void* d_out, int out_size, void* d_ws, size_t ws_size,
                              hipStream_t stream) {
    (void)in_sizes; (void)n_in; (void)d_ws; (void)ws_size; (void)out_size;

    const f32x4* x = (const f32x4*)d_in[0];        // [32,256,256,8] f32
    // d_in[1] is fsq_levels == 5 (compile-time constant here)

    float* out = (float*)d_out;
    f32x4*        quant = (f32x4*)out;                         // 16,777,216 f32
    f32x4*        zhat  = (f32x4*)(out + NELEM);               // 16,777,216 f32
    unsigned int* usage = (unsigned int*)(out + 2u * NELEM);   // 390,625 i32 counts

    // d_out is poisoned (0xAA) before timing and not re-poisoned between replays:
    // zero the histogram every call (deterministic, graph-capturable).
    fsq_zero_usage<<<(NBINS + 255) / 256, 256, 0, stream>>>(usage);
    fsq_main<<<NPIX / 256, 256, 0, stream>>>(x, quant, zhat, usage);
}